// ContactNet_83743272337894
// MI455X (gfx1250) — compile-verified
//
#include <hip/hip_runtime.h>
#include <cmath>

// 128-bit vector type for global_store_b128 emission
typedef float v4f __attribute__((ext_vector_type(4)));

__device__ __forceinline__ float fast_rsqrt(float x) {
    return __builtin_amdgcn_rsqf(x);   // v_rsq_f32, ~1 ulp
}
__device__ __forceinline__ float fast_rcp(float x) {
    return __builtin_amdgcn_rcpf(x);   // v_rcp_f32, ~1 ulp
}

__global__ __launch_bounds__(256)
void contactnet_grasp_kernel(const float* __restrict__ pts,
                             const float* __restrict__ z1p,
                             const float* __restrict__ z2p,
                             const float* __restrict__ slog,
                             const float* __restrict__ wlog,
                             float* __restrict__ out_pts,   // [N*3]
                             float* __restrict__ out_grasp, // [N*16]
                             float* __restrict__ out_s,     // [N]
                             float* __restrict__ out_w,     // [N]
                             int N)
{
    int i = blockIdx.x * blockDim.x + threadIdx.x;
    if (i >= N) return;

    const size_t i3 = (size_t)i * 3;

    // ---- streamed loads (merge to global_load_b96 / b32) ----
    float px = pts[i3 + 0], py = pts[i3 + 1], pz = pts[i3 + 2];
    float ax = z1p[i3 + 0], ay = z1p[i3 + 1], az = z1p[i3 + 2];
    float bx = z2p[i3 + 0], by = z2p[i3 + 1], bz = z2p[i3 + 2];
    float sx = slog[i];
    float wx = wlog[i];

    // base_dirs = z1 * rsqrt(dot(z1,z1))
    float inv1 = fast_rsqrt(ax*ax + ay*ay + az*az);
    float ux = ax*inv1, uy = ay*inv1, uz = az*inv1;

    // inner = dot(base, z2); approach = (z2 - inner*base) * rsqrt(dot(z2,z2))
    float inner = ux*bx + uy*by + uz*bz;
    float inv2  = fast_rsqrt(bx*bx + by*by + bz*bz);
    float apx = (bx - inner*ux) * inv2;
    float apy = (by - inner*uy) * inv2;
    float apz = (bz - inner*uz) * inv2;

    // col_x = unit(base_dirs)  (faithful re-normalization)
    float invb = fast_rsqrt(ux*ux + uy*uy + uz*uz);
    float cxx = ux*invb, cxy = uy*invb, cxz = uz*invb;

    // col_z = unit(approach_dirs)
    float inva = fast_rsqrt(apx*apx + apy*apy + apz*apz);
    float czx = apx*inva, czy = apy*inva, czz = apz*inva;

    // grasp_y = cross(col_z, col_x); col_y = unit(grasp_y)
    float gyx = czy*cxz - czz*cxy;
    float gyy = czz*cxx - czx*cxz;
    float gyz = czx*cxy - czy*cxx;
    float invy = fast_rsqrt(gyx*gyx + gyy*gyy + gyz*gyz);
    float cyx = gyx*invy, cyy = gyy*invy, cyz = gyz*invy;

    // sigmoid heads: 1/(1+exp(-x)) via v_exp + v_rcp
    float sv = fast_rcp(1.0f + __expf(-sx));
    float wv = fast_rcp(1.0f + __expf(-wx));

    // ---- outputs: non-temporal streaming stores ----
    // points (copy of contact_pts)
    __builtin_nontemporal_store(px, out_pts + i3 + 0);
    __builtin_nontemporal_store(py, out_pts + i3 + 1);
    __builtin_nontemporal_store(pz, out_pts + i3 + 2);

    // grasp[4][4], row-major: row r = [col_x[r], col_y[r], col_z[r], p[r]]
    v4f r0 = { cxx, cyx, czx, px };
    v4f r1 = { cxy, cyy, czy, py };
    v4f r2 = { cxz, cyz, czz, pz };
    v4f r3 = { 0.f, 0.f, 0.f, 1.f };
    v4f* g = (v4f*)(out_grasp + (size_t)i * 16);   // 16B-aligned (64 B/point)
    __builtin_nontemporal_store(r0, g + 0);
    __builtin_nontemporal_store(r1, g + 1);
    __builtin_nontemporal_store(r2, g + 2);
    __builtin_nontemporal_store(r3, g + 3);

    __builtin_nontemporal_store(sv, out_s + i);
    __builtin_nontemporal_store(wv, out_w + i);
}

extern "C" void kernel_launch(void* const* d_in, const int* in_sizes, int n_in,
                              void* d_out, int out_size, void* d_ws, size_t ws_size,
                              hipStream_t stream) {
    const float* pts = (const float*)d_in[0];   // [N,3]
    const float* z1  = (const float*)d_in[1];   // [N,3]
    const float* z2  = (const float*)d_in[2];   // [N,3]
    const float* sl  = (const float*)d_in[3];   // [N,1]
    const float* wl  = (const float*)d_in[4];   // [N,1]
    // d_in[5] = num_batches: only affects reshape; flat layout is identical.

    const int N = in_sizes[0] / 3;

    float* out      = (float*)d_out;
    float* out_pts  = out;                       // N*3
    float* out_g    = out + (size_t)N * 3;       // N*16
    float* out_s    = out + (size_t)N * 19;      // N
    float* out_w    = out + (size_t)N * 20;      // N

    const int threads = 256;                     // 8 wave32 waves per block
    const int blocks  = (N + threads - 1) / threads;
    contactnet_grasp_kernel<<<blocks, threads, 0, stream>>>(
        pts, z1, z2, sl, wl, out_pts, out_g, out_s, out_w, N);
}